// FeatureRelationDecoderV2_72834055406377
// MI455X (gfx1250) — compile-verified
//
#include <hip/hip_runtime.h>

typedef float v2f __attribute__((ext_vector_type(2)));
typedef float v4f __attribute__((ext_vector_type(4)));
typedef float v8f __attribute__((ext_vector_type(8)));

#define NN 3072
#define RR 16
#define TILES_PER_DIM (NN / 16)   // 192
#define WAVES_PER_BLOCK 8

// key[i] = batch[i] (>=0) if node i has a valid class, else a unique negative.
// Then pair-valid(i,j) (excluding seg/diag tests) <=> key[i] == key[j].
__global__ void frd_prep_key(const int* __restrict__ cls,
                             const int* __restrict__ batch,
                             int* __restrict__ key) {
  int i = blockIdx.x * blockDim.x + threadIdx.x;
  if (i < NN) {
    int c = cls[i];
    bool valid = (c != 24) && (c != 25) && (c != 26);
    key[i] = valid ? batch[i] : -(i + 1);
  }
}

__global__ __launch_bounds__(256) void frd_main(
    const float* __restrict__ z1, const float* __restrict__ z2,
    const float* __restrict__ seg, const int* __restrict__ key,
    float* __restrict__ out) {
  const int lane = threadIdx.x & 31;
  const int wave = (blockIdx.x * WAVES_PER_BLOCK) + (threadIdx.x >> 5);
  const int ti = wave / TILES_PER_DIM;
  const int tj = wave - ti * TILES_PER_DIM;
  const int i0 = ti * 16;
  const int j0 = tj * 16;

  const int n = lane & 15;       // column within tile (D-matrix N index)
  const int half = lane >> 4;    // 0: m rows 0..7, 1: m rows 8..15
  const int mBase = half * 8;

  const int jn = j0 + n;
  const int keyJ = key[jn];

  // A-operand lane owns M = lane&15; B-operand lane owns N = lane&15.
  const float* z1row = z1 + (size_t)(i0 + n) * RR;
  const float* z2row = z2 + (size_t)jn * RR;

  // Pair mask for this lane's 8 output rows (m = mBase..mBase+7), col n.
  unsigned mask = 0;
  #pragma unroll
  for (int v = 0; v < 8; ++v) {
    const int ig = i0 + mBase + v;
    const float s = seg[(size_t)ig * NN + jn];
    const bool p = (key[ig] == keyJ) & (s == 0.0f) & (ig != jn);
    mask |= (p ? 1u : 0u) << v;
  }

  // r handled in groups of 4: 4 live WMMA accumulators (32 VGPRs).
  #pragma unroll
  for (int g = 0; g < 4; ++g) {
    const int r0 = g * 4;
    const v4f a4 = *(const v4f*)(z1row + r0);
    const v4f b4 = *(const v4f*)(z2row + r0);
    const float av[4] = {a4.x, a4.y, a4.z, a4.w};
    const float bv[4] = {b4.x * 0.25f, b4.y * 0.25f, b4.z * 0.25f, b4.w * 0.25f};

    v8f acc[4];
    #pragma unroll
    for (int r = 0; r < 4; ++r) {
      // Replicate across all K slots: D = sum_k z1 * (z2/4) = z1*z2 exactly.
      v2f A = {av[r], av[r]};
      v2f B = {bv[r], bv[r]};
      v8f c = {};
      acc[r] = __builtin_amdgcn_wmma_f32_16x16x4_f32(
          /*neg_a=*/false, A, /*neg_b=*/false, B,
          /*c_mod=*/(short)0, c, /*reuse_a=*/false, /*reuse_b=*/false);
    }

    // Blend with default one-hot and stream out (non-temporal B128 stores).
    #pragma unroll
    for (int v = 0; v < 8; ++v) {
      const int ig = i0 + mBase + v;
      const bool p = (mask >> v) & 1u;
      v4f o;
      if (p) {
        o.x = acc[0][v]; o.y = acc[1][v]; o.z = acc[2][v]; o.w = acc[3][v];
      } else {
        o.x = (r0 == 0) ? 1.0f : 0.0f;
        o.y = 0.0f; o.z = 0.0f; o.w = 0.0f;
      }
      v4f* dst = (v4f*)(out + ((size_t)ig * NN + jn) * RR + r0);
      __builtin_nontemporal_store(o, dst);
    }
  }
}

extern "C" void kernel_launch(void* const* d_in, const int* in_sizes, int n_in,
                              void* d_out, int out_size, void* d_ws, size_t ws_size,
                              hipStream_t stream) {
  const float* z1  = (const float*)d_in[0];
  const float* z2  = (const float*)d_in[1];
  const float* seg = (const float*)d_in[2];
  const int* cls   = (const int*)d_in[3];
  const int* batch = (const int*)d_in[4];
  float* out = (float*)d_out;
  int* key = (int*)d_ws;   // NN ints

  frd_prep_key<<<(NN + 255) / 256, 256, 0, stream>>>(cls, batch, key);

  const int total_tiles = TILES_PER_DIM * TILES_PER_DIM;          // 36864
  const int blocks = total_tiles / WAVES_PER_BLOCK;               // 4608
  frd_main<<<blocks, 256, 0, stream>>>(z1, z2, seg, key, out);
}